// CRF_53592601919901
// MI455X (gfx1250) — compile-verified
//
#include <hip/hip_runtime.h>
#include <math.h>

// Problem constants (fixed by the reference setup_inputs()).
#define NUM_TAGS 48
#define S        50      // NUM_TAGS + START + END
#define P        64      // padded state count (WMMA N/K tiles)
#define WSTR     66      // LDS state row stride (even -> 8B-aligned b64 A reads)
#define EMSTR    48      // packed emission tile row stride (TDM-friendly)
#define BATCH    512
#define SEQ      1024
#define ROWS     16      // batch rows per workgroup (WMMA M)
#define TPB      128     // 4 waves * 32 lanes

typedef float v2f __attribute__((ext_vector_type(2)));
typedef float v8f __attribute__((ext_vector_type(8)));
typedef unsigned int u32x4 __attribute__((ext_vector_type(4)));
typedef int i32x4 __attribute__((ext_vector_type(4)));
typedef int i32x8 __attribute__((ext_vector_type(8)));

#ifndef HAVE_TDM
#if defined(__has_builtin)
#if __has_builtin(__builtin_amdgcn_tensor_load_to_lds)
#define HAVE_TDM 1
#endif
#endif
#endif
#ifndef HAVE_TDM
#define HAVE_TDM 0
#endif

__device__ __forceinline__ int rfl(int x) { return __builtin_amdgcn_readfirstlane(x); }

#if HAVE_TDM
// 2D TDM: load a 48-wide x nrows-tall fp32 tile (row pitch SEQ*NUM_TAGS elems)
// from global into LDS at byte offset lds_off (rows packed at 192B).
__device__ __forceinline__ void tdm_load_rows4(const float* gsrc, unsigned lds_off) {
  unsigned long long ga = (unsigned long long)(uintptr_t)gsrc;
  u32x4 g0;
  g0[0] = (unsigned)rfl(1);                                   // count=1, user mode
  g0[1] = (unsigned)rfl((int)lds_off);                        // LDS byte address
  g0[2] = (unsigned)rfl((int)(unsigned)(ga & 0xffffffffull)); // global_addr lo
  g0[3] = (unsigned)rfl((int)(((unsigned)(ga >> 32) & 0x01ffffffu) | (2u << 30))); // hi | type=2
  i32x8 g1;
  g1[0] = rfl(0x00020000);              // workgroup_mask=0, data_size=2 (4B)
  g1[1] = rfl((int)(48u << 16));        // tensor_dim0 = 48  (bits 63:48)
  g1[2] = rfl((int)(4u << 16));         // tensor_dim1 = 4   (bits 95:80)
  g1[3] = rfl((int)(48u << 16));        // tile_dim0 = 48    (bits 127:112)
  g1[4] = rfl(4);                       // tile_dim1 = 4     (bits 143:128)
  g1[5] = rfl(SEQ * NUM_TAGS);          // tensor_dim0_stride lo32 (bits 191:160)
  g1[6] = 0;
  g1[7] = 0;
  i32x4 gz = {0, 0, 0, 0};
#if __clang_major__ >= 23
  i32x8 gz8 = {0, 0, 0, 0, 0, 0, 0, 0};
  __builtin_amdgcn_tensor_load_to_lds(g0, g1, gz, gz, gz8, 0);
#else
  __builtin_amdgcn_tensor_load_to_lds(g0, g1, gz, gz, 0);
#endif
}

__device__ __forceinline__ void wait_tensor0() {
#if defined(__has_builtin) && __has_builtin(__builtin_amdgcn_s_wait_tensorcnt)
  __builtin_amdgcn_s_wait_tensorcnt(0);
#else
  asm volatile("s_wait_tensorcnt 0x0" ::: "memory");
#endif
}
#endif // HAVE_TDM

// ---------------------------------------------------------------------------
// Forward (log-partition): scaled-linear-space CRF forward recursion.
//   w' = (w @ exp(T)) * exp(emit),  m += log(rowmax) every 4 steps
//   log_den = m + log(sum_j w_j * exp(T[j, END]))
// One WG = 16 batch rows; wave wv owns output columns [16*wv, 16*wv+16).
// ---------------------------------------------------------------------------
__global__ __launch_bounds__(TPB) void crf_forward_wmma(
    const float* __restrict__ emis, const float* __restrict__ mask,
    const float* __restrict__ trans, float* __restrict__ logden)
{
  __shared__ float E[P * WSTR];            // exp(transitions), zero padded
  __shared__ float Wb[2][ROWS * WSTR];     // ping-pong state (scaled linear)
  __shared__ float EM[2][ROWS * EMSTR];    // raw emissions double buffer
  __shared__ float MB[2][ROWS];            // mask double buffer
  __shared__ float MROW[ROWS];             // per-row log scale m
  __shared__ float RED[ROWS * 8];          // renorm partial maxes
  __shared__ float INV[ROWS];              // renorm reciprocal scales
  __shared__ float EEND[P];                // exp(T[:, END])

  const int tid   = threadIdx.x;
  const int lane  = tid & 31;
  const int wv    = tid >> 5;
  const int b0    = blockIdx.x * ROWS;
  const int nn    = wv * 16 + (lane & 15);   // this lane's column
  const int khalf = (lane >> 4) * 2;         // f32 A/B frag K sub-offset
  const int rbase = (lane >> 4) * 8;         // f32 C/D frag row base
  const int colr  = (nn < NUM_TAGS) ? nn : (NUM_TAGS - 1); // clamped emit col
  const bool realcol = (nn < NUM_TAGS);

  // ---- init E = exp(T) (NEG_INF -> 0 kills START-in / END-out arcs) ----
  for (int e = tid; e < P * P; e += TPB) {
    int i = e >> 6, j = e & 63;
    float v = 0.0f;
    if (i < S && j < S) v = __expf(trans[i * S + j]);
    E[i * WSTR + j] = v;
  }
  for (int j = tid; j < P; j += TPB)
    EEND[j] = (j < S) ? __expf(trans[j * S + (S - 1)]) : 0.0f;
  for (int e = tid; e < ROWS * P; e += TPB) {
    int r = e >> 6, c = e & 63;
    Wb[0][r * WSTR + c] = (c == NUM_TAGS) ? 1.0f : 0.0f;   // one-hot START
  }
  if (tid < ROWS) MROW[tid] = 0.0f;

  // ---- preload t=0 emissions + mask ----
#if HAVE_TDM
  const unsigned em_lds0 = (unsigned)(uintptr_t)&EM[0][0];
  const unsigned em_lds1 = (unsigned)(uintptr_t)&EM[1][0];
  tdm_load_rows4(emis + ((size_t)(b0 + wv * 4) * SEQ) * NUM_TAGS,
                 em_lds0 + (unsigned)(wv * 4 * EMSTR * 4));
#else
  int er[6], ec[6];
  #pragma unroll
  for (int k = 0; k < 6; k++) {
    int e = tid + k * TPB; er[k] = e / NUM_TAGS; ec[k] = e % NUM_TAGS;
  }
  #pragma unroll
  for (int k = 0; k < 6; k++)
    EM[0][tid + k * TPB] = emis[((b0 + er[k]) * SEQ + 0) * NUM_TAGS + ec[k]];
#endif
  if (tid < ROWS) MB[0][tid] = mask[(b0 + tid) * SEQ + 0];
#if HAVE_TDM
  wait_tensor0();
#endif
  __syncthreads();

  // ---- resident B fragments: E[k, nn] for all K, kept in VGPRs ----
  v2f bfr[16];
  #pragma unroll
  for (int f = 0; f < 16; f++) {
    int k0 = 4 * f + khalf;
    bfr[f].x = E[k0 * WSTR + nn];
    bfr[f].y = E[(k0 + 1) * WSTR + nn];
  }

  for (int t = 0; t < SEQ; t++) {
    const int cur = t & 1, nxt = cur ^ 1;
    const float* Wc  = Wb[cur];
    float*       Wn  = Wb[nxt];
    const float* EMc = EM[cur];

    // -- kick off t+1 emission DMA / prefetch + mask (overlaps WMMA) --
    const int tn = (t + 1 < SEQ) ? (t + 1) : (SEQ - 1);
#if HAVE_TDM
    tdm_load_rows4(emis + ((size_t)(b0 + wv * 4) * SEQ + tn) * NUM_TAGS,
                   (nxt ? em_lds1 : em_lds0) + (unsigned)(wv * 4 * EMSTR * 4));
#else
    float pre[6];
    #pragma unroll
    for (int k = 0; k < 6; k++)
      pre[k] = emis[((b0 + er[k]) * SEQ + tn) * NUM_TAGS + ec[k]];
#endif
    float mpre = 0.0f;
    if (tid < ROWS) mpre = mask[(b0 + tid) * SEQ + tn];

    // -- per-step mask summary (wave-uniform fast path) --
    float mrow_l = MB[cur][lane & 15];
    const bool allm =
        (((unsigned long long)__ballot(mrow_l > 0.0f)) & 0xffffffffull) == 0xffffffffull;

    // -- D = W @ E : K=64 as two independent 8-deep WMMA chains --
    v8f acc0 = {0.f, 0.f, 0.f, 0.f, 0.f, 0.f, 0.f, 0.f};
    v8f acc1 = {0.f, 0.f, 0.f, 0.f, 0.f, 0.f, 0.f, 0.f};
    #pragma unroll
    for (int f = 0; f < 8; f++) {
      int k0 = 4 * f + khalf;
      v2f a = *(const v2f*)&Wc[(lane & 15) * WSTR + k0];
      acc0 = __builtin_amdgcn_wmma_f32_16x16x4_f32(
          false, a, false, bfr[f], (short)0, acc0, false, false);
    }
    #pragma unroll
    for (int f = 8; f < 16; f++) {
      int k0 = 4 * f + khalf;
      v2f a = *(const v2f*)&Wc[(lane & 15) * WSTR + k0];
      acc1 = __builtin_amdgcn_wmma_f32_16x16x4_f32(
          false, a, false, bfr[f], (short)0, acc1, false, false);
    }

    // -- apply exp(emissions), branch-free pad handling, store W_next --
    #pragma unroll
    for (int k = 0; k < 8; k++) {
      int row = rbase + k;
      float ex  = __expf(EMc[row * EMSTR + colr]);
      float emv = realcol ? ex : 1.0f;           // pad cols: emit_ext = 0
      Wn[row * WSTR + nn] = (acc0[k] + acc1[k]) * emv;
    }
    if (!allm) {                                  // rare general-mask patch-up
      #pragma unroll
      for (int k = 0; k < 8; k++) {
        int row = rbase + k;
        if (MB[cur][row] <= 0.0f)
          Wn[row * WSTR + nn] = Wc[row * WSTR + nn];
      }
    }

    // -- publish prefetched emissions (fallback only) + mask --
#if !HAVE_TDM
    #pragma unroll
    for (int k = 0; k < 6; k++) EM[nxt][tid + k * TPB] = pre[k];
#endif
    if (tid < ROWS) MB[nxt][tid] = mpre;

#if HAVE_TDM
    wait_tensor0();          // t+1 tile landed before we barrier
#endif
    __syncthreads();

    // -- per-row renormalization every 4 steps (keeps fp32 in range) --
    if ((t & 3) == 3) {
      {
        int row = tid >> 3, seg = tid & 7;
        float mx = 0.0f;
        const float* wr = &Wn[row * WSTR + seg * 8];
        #pragma unroll
        for (int q = 0; q < 8; q++) mx = fmaxf(mx, wr[q]);
        RED[row * 8 + seg] = mx;
      }
      __syncthreads();
      if (tid < ROWS) {
        float mx = 0.0f;
        #pragma unroll
        for (int q = 0; q < 8; q++) mx = fmaxf(mx, RED[tid * 8 + q]);
        if (mx > 0.0f) { MROW[tid] += __logf(mx); INV[tid] = 1.0f / mx; }
        else           { INV[tid] = 1.0f; }
      }
      __syncthreads();
      for (int e = tid; e < ROWS * P; e += TPB) {
        int r = e >> 6, c = e & 63;
        Wn[r * WSTR + c] *= INV[r];
      }
      __syncthreads();
    }
  }

  // ---- finalize: log_den = m + log(sum_j w_j * exp(T[j,END])) ----
  const int fin = ((SEQ - 1) & 1) ^ 1;
  if (tid < ROWS) {
    float s = 0.0f;
    for (int j = 0; j < P; j++) s += Wb[fin][tid * WSTR + j] * EEND[j];
    s = fmaxf(s, 1e-37f);
    logden[b0 + tid] = MROW[tid] + __logf(s);
  }
}

// ---------------------------------------------------------------------------
// Gold path score: one thread per batch row, sequential gather-scan.
// ---------------------------------------------------------------------------
__global__ __launch_bounds__(TPB) void crf_gold(
    const float* __restrict__ emis, const int* __restrict__ tags,
    const float* __restrict__ mask, const float* __restrict__ trans,
    float* __restrict__ lognum)
{
  int b = blockIdx.x * blockDim.x + threadIdx.x;
  if (b >= BATCH) return;
  float score = 0.0f;
  int prev = NUM_TAGS;                        // START
  const int baseT = b * SEQ;
  for (int t = 0; t < SEQ; t++) {
    int   cur = tags[baseT + t];
    float mv  = mask[baseT + t];
    float em  = (cur < NUM_TAGS) ? emis[(baseT + t) * NUM_TAGS + cur] : 0.0f;
    float tr  = trans[prev * S + cur];
    score += (em + tr) * mv;
    prev = (mv > 0.0f) ? cur : prev;
  }
  score += trans[prev * S + (S - 1)];         // -> END
  lognum[b] = score;
}

// ---------------------------------------------------------------------------
// Final scalar: mean(log_den - log_num)
// ---------------------------------------------------------------------------
__global__ __launch_bounds__(256) void crf_reduce(
    const float* __restrict__ den, const float* __restrict__ num,
    float* __restrict__ out)
{
  __shared__ float red[256];
  int tid = threadIdx.x;
  float s = 0.0f;
  for (int i = tid; i < BATCH; i += 256) s += den[i] - num[i];
  red[tid] = s;
  __syncthreads();
  for (int w = 128; w > 0; w >>= 1) {
    if (tid < w) red[tid] += red[tid + w];
    __syncthreads();
  }
  if (tid == 0) out[0] = red[0] / (float)BATCH;
}

extern "C" void kernel_launch(void* const* d_in, const int* in_sizes, int n_in,
                              void* d_out, int out_size, void* d_ws, size_t ws_size,
                              hipStream_t stream) {
  (void)in_sizes; (void)n_in; (void)out_size; (void)ws_size;
  const float* emis  = (const float*)d_in[0];   // (B, L, C) fp32
  const int*   tags  = (const int*)  d_in[1];   // (B, L) int32
  const float* maskp = (const float*)d_in[2];   // (B, L) fp32
  const float* trans = (const float*)d_in[3];   // (C+2, C+2) fp32
  float* out = (float*)d_out;

  float* den = (float*)d_ws;                    // [BATCH]
  float* num = den + BATCH;                     // [BATCH]

  crf_forward_wmma<<<dim3(BATCH / ROWS), dim3(TPB), 0, stream>>>(emis, maskp, trans, den);
  crf_gold<<<dim3((BATCH + TPB - 1) / TPB), dim3(TPB), 0, stream>>>(emis, tags, maskp, trans, num);
  crf_reduce<<<dim3(1), dim3(256), 0, stream>>>(den, num, out);
}